// Attention_17978733101505
// MI455X (gfx1250) — compile-verified
//
#include <hip/hip_runtime.h>
#include <math.h>

#define B_SZ   64
#define T_SZ   2048
#define H_SZ   512
#define U_SZ   128
#define ROWS   64                    // timesteps per chunk
#define NCHUNK (T_SZ / ROWS)         // 32

typedef __attribute__((ext_vector_type(2))) float v2f;
typedef __attribute__((ext_vector_type(8))) float v8f;

// ---------------------------------------------------------------------------
// Kernel A: v[b,h] = sum_k W_score[h,k] * h_t[b,k]   (h_t = hidden[:, T-1, :])
// One wave per 16x16 tile of V (M=batch=64, N=h=512, K=512).
// ---------------------------------------------------------------------------
__global__ void score_vec_kernel(const float* __restrict__ hid,
                                 const float* __restrict__ Wscore,
                                 float* __restrict__ v) {
  const int lane = threadIdx.x;          // 32 threads = 1 wave
  const int bm   = blockIdx.x >> 5;      // batch tile 0..3
  const int bn   = blockIdx.x & 31;      // h tile 0..31
  const int half = lane >> 4;            // 0/1
  const int l16  = lane & 15;
  const int koff = half * 2;

  // A[m][k] = h_t[bm*16+m][k]
  const float* arow = hid + ((size_t)(bm * 16 + l16) * T_SZ + (T_SZ - 1)) * H_SZ;
  // B[k][n] = W_score[bn*16+n][k]  (row of W_score, contiguous in k)
  const float* brow = Wscore + (size_t)(bn * 16 + l16) * H_SZ;

  v8f c = {};
  for (int k0 = 0; k0 < H_SZ; k0 += 4) {
    v2f a, bb;
    a.x  = arow[k0 + koff];
    a.y  = arow[k0 + koff + 1];
    bb.x = brow[k0 + koff];
    bb.y = brow[k0 + koff + 1];
    c = __builtin_amdgcn_wmma_f32_16x16x4_f32(false, a, false, bb,
                                              (short)0, c, false, false);
  }
#pragma unroll
  for (int r = 0; r < 8; ++r) {
    const int m = bm * 16 + r + 8 * half;
    const int n = bn * 16 + l16;
    v[m * H_SZ + n] = c[r];
  }
}

// ---------------------------------------------------------------------------
// Kernel B: per (batch, 64-row chunk): stage chunk into LDS with CDNA5 async
// global->LDS copies (no VGPR round-trip, ASYNCcnt-tracked), then scores +
// chunk-local softmax partials + partial context from the same LDS tile.
// blockDim = 256 (8 waves). Dynamic LDS: 64*512 + 2*64 floats = 128.5 KB.
// ---------------------------------------------------------------------------
__global__ void attn_partial_kernel(const float* __restrict__ hid,
                                    const float* __restrict__ v,
                                    float* __restrict__ chunk_m,
                                    float* __restrict__ chunk_l,
                                    float* __restrict__ chunk_ctx) {
  extern __shared__ float smem[];
  float* tile   = smem;                  // ROWS * H_SZ
  float* scores = smem + ROWS * H_SZ;    // ROWS
  float* pbuf   = scores + ROWS;         // ROWS

  const int tid  = threadIdx.x;
  const int wave = tid >> 5;
  const int lane = tid & 31;
  const int cc   = blockIdx.x;           // chunk
  const int b    = blockIdx.y;           // batch
  const int t0   = cc * ROWS;

  // Stage: rows [t0, t0+64) of hidden[b] are one contiguous 128 KB block.
  // Async global->LDS b128 copies; dynamic LDS begins at groupstaticsize().
  {
    const float4* src = (const float4*)(hid + ((size_t)b * T_SZ + t0) * H_SZ);
    const unsigned lds0 = __builtin_amdgcn_groupstaticsize();
    for (int i = tid; i < ROWS * H_SZ / 4; i += 256) {
      const unsigned loff = lds0 + (unsigned)i * 16u;
      const float4* g = src + i;
      asm volatile("global_load_async_to_lds_b128 %0, %1, off"
                   :: "v"(loff), "v"(g)
                   : "memory");
    }
  }

  // Overlap with the async copy: each lane caches its slice of v[b]
  // (elements i*128 + lane*4 .. +3, i = 0..3).
  const float4* v4 = (const float4*)(v + b * H_SZ);
  const float4 vr0 = v4[lane +  0];
  const float4 vr1 = v4[lane + 32];
  const float4 vr2 = v4[lane + 64];
  const float4 vr3 = v4[lane + 96];

  asm volatile("s_wait_asynccnt 0" ::: "memory");
  __syncthreads();

  // Wave-cooperative dot products: score[t] = tile[t,:] . v[b,:]
  for (int r = wave; r < ROWS; r += 8) {
    const float4* hrow = (const float4*)(tile + r * H_SZ);
    const float4 h0 = hrow[lane +  0];
    const float4 h1 = hrow[lane + 32];
    const float4 h2 = hrow[lane + 64];
    const float4 h3 = hrow[lane + 96];
    float s = h0.x * vr0.x + h0.y * vr0.y + h0.z * vr0.z + h0.w * vr0.w
            + h1.x * vr1.x + h1.y * vr1.y + h1.z * vr1.z + h1.w * vr1.w
            + h2.x * vr2.x + h2.y * vr2.y + h2.z * vr2.z + h2.w * vr2.w
            + h3.x * vr3.x + h3.y * vr3.y + h3.z * vr3.z + h3.w * vr3.w;
#pragma unroll
    for (int off = 16; off > 0; off >>= 1) s += __shfl_xor(s, off, 32);
    if (lane == 0) scores[r] = s;
  }
  __syncthreads();

  // Chunk-local max / exp / sum (redundant per thread; trivially cheap)
  float m = -INFINITY;
  for (int t = 0; t < ROWS; ++t) m = fmaxf(m, scores[t]);
  if (tid < ROWS) pbuf[tid] = __expf(scores[tid] - m);
  __syncthreads();
  float l = 0.f;
  for (int t = 0; t < ROWS; ++t) l += pbuf[t];

  // Partial context from the SAME LDS tile (no second HBM read)
  float* ctxout = chunk_ctx + (size_t)(b * NCHUNK + cc) * H_SZ;
  for (int h = tid; h < H_SZ; h += 256) {
    float acc = 0.f;
    for (int t = 0; t < ROWS; ++t) acc += pbuf[t] * tile[t * H_SZ + h];
    ctxout[h] = acc;
  }
  if (tid == 0) {
    chunk_m[b * NCHUNK + cc] = m;
    chunk_l[b * NCHUNK + cc] = l;
  }
}

// ---------------------------------------------------------------------------
// Kernel C: merge chunk partials (log-sum-exp rescale), build pre=[ctx, h_t]
// ---------------------------------------------------------------------------
__global__ void finalize_ctx_kernel(const float* __restrict__ hid,
                                    const float* __restrict__ chunk_m,
                                    const float* __restrict__ chunk_l,
                                    const float* __restrict__ chunk_ctx,
                                    float* __restrict__ pre) {
  const int b   = blockIdx.x;
  const int tid = threadIdx.x;

  float M = -INFINITY;
  for (int c = 0; c < NCHUNK; ++c) M = fmaxf(M, chunk_m[b * NCHUNK + c]);
  float L = 0.f;
  for (int c = 0; c < NCHUNK; ++c)
    L += chunk_l[b * NCHUNK + c] * __expf(chunk_m[b * NCHUNK + c] - M);
  const float invL = 1.0f / L;

  for (int h = tid; h < H_SZ; h += 256) {
    float acc = 0.f;
    for (int c = 0; c < NCHUNK; ++c)
      acc += __expf(chunk_m[b * NCHUNK + c] - M) *
             chunk_ctx[(size_t)(b * NCHUNK + c) * H_SZ + h];
    pre[b * 2 * H_SZ + h] = acc * invL;
  }
  const float* ht = hid + ((size_t)b * T_SZ + (T_SZ - 1)) * H_SZ;
  for (int h = tid; h < H_SZ; h += 256)
    pre[b * 2 * H_SZ + H_SZ + h] = ht[h];
}

// ---------------------------------------------------------------------------
// Kernel D: out = tanh(pre @ W_out), pre (64x1024), W_out (1024x128).
// One wave per 16x16 output tile; WMMA f32 16x16x4, K-loop over 1024.
// ---------------------------------------------------------------------------
__global__ void out_gemm_kernel(const float* __restrict__ pre,
                                const float* __restrict__ Wout,
                                float* __restrict__ out) {
  const int lane = threadIdx.x;
  const int bm   = blockIdx.x >> 3;      // batch tile 0..3
  const int bn   = blockIdx.x & 7;       // unit tile 0..7
  const int half = lane >> 4;
  const int l16  = lane & 15;
  const int koff = half * 2;

  const float* arow = pre + (size_t)(bm * 16 + l16) * (2 * H_SZ);
  const float* bcol = Wout + (bn * 16 + l16);   // B[k][n] = Wout[k*128 + n]

  v8f c = {};
  for (int k0 = 0; k0 < 2 * H_SZ; k0 += 4) {
    v2f a, bb;
    a.x  = arow[k0 + koff];
    a.y  = arow[k0 + koff + 1];
    bb.x = bcol[(size_t)(k0 + koff) * U_SZ];
    bb.y = bcol[(size_t)(k0 + koff + 1) * U_SZ];
    c = __builtin_amdgcn_wmma_f32_16x16x4_f32(false, a, false, bb,
                                              (short)0, c, false, false);
  }
#pragma unroll
  for (int r = 0; r < 8; ++r) {
    const int m = bm * 16 + r + 8 * half;
    const int n = bn * 16 + l16;
    out[m * U_SZ + n] = tanhf(c[r]);
  }
}

// ---------------------------------------------------------------------------
extern "C" void kernel_launch(void* const* d_in, const int* in_sizes, int n_in,
                              void* d_out, int out_size, void* d_ws, size_t ws_size,
                              hipStream_t stream) {
  const float* hid    = (const float*)d_in[0];   // (64, 2048, 512) f32
  const float* Wscore = (const float*)d_in[1];   // (512, 512) f32
  const float* Wout   = (const float*)d_in[2];   // (1024, 128) f32
  float* out = (float*)d_out;                    // (64, 128) f32

  // Workspace layout (floats): v | chunk_m | chunk_l | chunk_ctx | pre
  float* ws        = (float*)d_ws;
  float* v         = ws;                                   // B*H      = 32768
  float* chunk_m   = v + B_SZ * H_SZ;                      // B*NCHUNK = 2048
  float* chunk_l   = chunk_m + B_SZ * NCHUNK;              // 2048
  float* chunk_ctx = chunk_l + B_SZ * NCHUNK;              // B*NCHUNK*H = 1048576
  float* pre       = chunk_ctx + (size_t)B_SZ * NCHUNK * H_SZ; // B*1024 = 65536

  // 1) v = h_t @ W_score^T  (WMMA f32)
  score_vec_kernel<<<dim3(128), dim3(32), 0, stream>>>(hid, Wscore, v);

  // 2) single streaming pass over hidden: chunk softmax partials + context
  const size_t smem = (size_t)(ROWS * H_SZ + 2 * ROWS) * sizeof(float); // ~128.5 KB (<320 KB/WGP)
  attn_partial_kernel<<<dim3(NCHUNK, B_SZ), dim3(256), smem, stream>>>(
      hid, v, chunk_m, chunk_l, chunk_ctx);

  // 3) merge chunks, assemble pre = [context, h_t]
  finalize_ctx_kernel<<<dim3(B_SZ), dim3(256), 0, stream>>>(
      hid, chunk_m, chunk_l, chunk_ctx, pre);

  // 4) out = tanh(pre @ W_out)  (WMMA f32)
  out_gemm_kernel<<<dim3(32), dim3(32), 0, stream>>>(pre, Wout, out);
}